// QLayerChunked_45878840655941
// MI455X (gfx1250) — compile-verified
//
#include <hip/hip_runtime.h>
#include <cstdint>

// ---------------------------------------------------------------------------
// rho_out = V rho V^dagger, V = U_15 ... U_0 (SU(2)).
// Euler form U_k = Z(p1)Ry(p0)Z(p2) telescopes: adjacent Z phases merge,
// so only 33 sincos pairs per element (66 trans ops) instead of 48 (96).
// Per-block constants (incl. the 1/(4*pi) hardware-sin scale) are folded once
// per workgroup into LDS. Memory-bound: 256 MiB @ 23.3 TB/s => ~11 us floor.
// CDNA5 paths: global_load_async_to_lds_b32 + s_wait_asynccnt, global_prefetch,
// non-temporal b128 loads/stores.
// ---------------------------------------------------------------------------

typedef float v4f __attribute__((ext_vector_type(4)));

#define INV4PI 0.07957747154594767f   // 1/(4*pi): half-angle * 1/(2*pi)

// hardware sin/cos take input in revolutions (sin(2*pi*x))
__device__ __forceinline__ float hwsin(float cyc) {
#if __has_builtin(__builtin_amdgcn_sinf)
    return __builtin_amdgcn_sinf(cyc);
#else
    return __sinf(cyc * 6.2831853071795864f);
#endif
}
__device__ __forceinline__ float hwcos(float cyc) {
#if __has_builtin(__builtin_amdgcn_cosf)
    return __builtin_amdgcn_cosf(cyc);
#else
    return __cosf(cyc * 6.2831853071795864f);
#endif
}

struct c32 { float re, im; };
__device__ __forceinline__ c32 cmul(c32 a, c32 b) {          // a*b
    return { fmaf(a.re, b.re, -a.im * b.im),
             fmaf(a.re, b.im,  a.im * b.re) };
}
__device__ __forceinline__ c32 cmulc(c32 a, c32 b) {         // a*conj(b)
    return { fmaf(a.re, b.re,  a.im * b.im),
             fmaf(a.im, b.re, -a.re * b.im) };
}
__device__ __forceinline__ c32 cadd(c32 a, c32 b) { return {a.re + b.re, a.im + b.im}; }

__global__ __launch_bounds__(256)
void qlayer_su2_kernel(const float* __restrict__ rho,    // [B,2,2] complex64 -> 8 f32/elem
                       const float* __restrict__ x,      // [B,48]
                       const float* __restrict__ wv,     // [48]
                       const float* __restrict__ th,     // [48]
                       float* __restrict__ out,          // [B,2,2] complex64 -> 8 f32/elem
                       int B)
{
    __shared__ float s_raw[96];   // [0..47] = w, [48..95] = theta
    __shared__ float s_c[96];     // per block k: {cw0, ct0, cw1, cw2n, ctm, pad}
    __shared__ float s_ci[2];     // init phase consts: {w[2]/(4pi), th[2]/(4pi)}

    const int t = threadIdx.x;

    // --- Stage raw w/theta via CDNA5 async-to-LDS (ASYNCcnt path) -----------
    if (t < 48) {
        uint32_t lw = (uint32_t)(uintptr_t)&s_raw[t];
        uint32_t lt = (uint32_t)(uintptr_t)&s_raw[48 + t];
        const float* gw = wv + t;
        const float* gt = th + t;
        asm volatile("global_load_async_to_lds_b32 %0, %1, off"
                     :: "v"(lw), "v"(gw) : "memory");
        asm volatile("global_load_async_to_lds_b32 %0, %1, off"
                     :: "v"(lt), "v"(gt) : "memory");
    }
    asm volatile("s_wait_asynccnt 0x0" ::: "memory");
    __syncthreads();

    // --- Fold per-block constants (one thread per block) --------------------
    if (t < 16) {
        const float w0  = s_raw[3*t + 0];
        const float w1  = s_raw[3*t + 1];
        const float w2n = (t < 15) ? s_raw[3*t + 5] : 0.0f;        // w[3(k+1)+2]
        const float t0  = s_raw[48 + 3*t + 0];
        const float t1  = s_raw[48 + 3*t + 1];
        const float t2n = (t < 15) ? s_raw[48 + 3*t + 5] : 0.0f;   // th[3(k+1)+2]
        s_c[6*t + 0] = w0 * INV4PI;
        s_c[6*t + 1] = t0 * INV4PI;
        s_c[6*t + 2] = w1 * INV4PI;
        s_c[6*t + 3] = w2n * INV4PI;
        s_c[6*t + 4] = (t1 + t2n) * INV4PI;
        s_c[6*t + 5] = 0.0f;
        if (t == 0) {
            s_ci[0] = s_raw[2]      * INV4PI;   // w[2]/(4pi)
            s_ci[1] = s_raw[48 + 2] * INV4PI;   // th[2]/(4pi)
        }
    }
    __syncthreads();

    const long long b = (long long)blockIdx.x * blockDim.x + threadIdx.x;
    if (b >= B) return;

    // Prefetch the x stream ahead (speculative; OOB translations dropped).
    __builtin_prefetch(x + (b + 12LL * 256) * 48, 0, 0);

    // --- Streaming (non-temporal) b128 loads --------------------------------
    const v4f* xv = (const v4f*)(x + b * 48);
    v4f xq[12];
#pragma unroll
    for (int i = 0; i < 12; ++i) xq[i] = __builtin_nontemporal_load(&xv[i]);
    const float* px = (const float*)xq;

    const v4f* rv = (const v4f*)(rho + b * 8);
    v4f r0 = __builtin_nontemporal_load(&rv[0]);
    v4f r1 = __builtin_nontemporal_load(&rv[1]);
    c32 R00{r0.x, r0.y}, R01{r0.z, r0.w}, R10{r1.x, r1.y}, R11{r1.z, r1.w};

    // --- Telescoped Euler chain: V = Z(m15) Ry(a15) ... Ry(a0) Z(p2_0) ------
    // State is SU(2): [[A,B],[-conj(B),conj(A)]]
    const float icyc = fmaf(s_ci[0], px[2], s_ci[1]);     // p2_0 / (4pi)
    float Are = hwcos(icyc), Aim = hwsin(icyc);
    float Bre = 0.0f,        Bim = 0.0f;

#pragma unroll
    for (int k = 0; k < 16; ++k) {
        const int i2n = (k < 15) ? (3*k + 5) : 0;         // coeff is 0 for k==15
        const float acyc = fmaf(s_c[6*k + 0], px[3*k + 0], s_c[6*k + 1]);
        const float mcyc = fmaf(s_c[6*k + 2], px[3*k + 1],
                           fmaf(s_c[6*k + 3], px[i2n],     s_c[6*k + 4]));
        const float sa = hwsin(acyc), ca = hwcos(acyc);
        // S <- Ry(a_k) * S : A' = ca*A + sa*conj(B); B' = ca*B - sa*conj(A)
        const float Ar = fmaf( sa, Bre, ca * Are);
        const float Ai = fmaf(-sa, Bim, ca * Aim);
        const float Br = fmaf(-sa, Are, ca * Bre);
        const float Bi = fmaf( sa, Aim, ca * Bim);
        const float sm = hwsin(mcyc), cm = hwcos(mcyc);
        // S <- Z(m_k) * S : A'' = e^{i m/2} A'; B'' = e^{i m/2} B'
        Are = fmaf(-sm, Ai, cm * Ar);
        Aim = fmaf( sm, Ar, cm * Ai);
        Bre = fmaf(-sm, Bi, cm * Br);
        Bim = fmaf( sm, Br, cm * Bi);
    }

    const c32 A {Are, Aim};
    const c32 Bb{Bre, Bim};

    // --- Single sandwich: out = V rho V^dagger ------------------------------
    const c32 cA { A.re, -A.im};              //  conj(A)
    const c32 nCB{-Bb.re,  Bb.im};            // -conj(B)
    const c32 nB {-Bb.re, -Bb.im};            // -B

    // M = V * rho
    const c32 M00 = cadd(cmul(A,   R00), cmul(Bb, R10));
    const c32 M01 = cadd(cmul(A,   R01), cmul(Bb, R11));
    const c32 M10 = cadd(cmul(nCB, R00), cmul(cA, R10));
    const c32 M11 = cadd(cmul(nCB, R01), cmul(cA, R11));

    // out = M * V^dagger, V^dagger = [[conj(A), -B], [conj(B), A]]
    const c32 O00 = cadd(cmulc(M00, A),  cmulc(M01, Bb));
    const c32 O01 = cadd(cmul (M00, nB), cmul (M01, A));
    const c32 O10 = cadd(cmulc(M10, A),  cmulc(M11, Bb));
    const c32 O11 = cadd(cmul (M10, nB), cmul (M11, A));

    v4f* ov = (v4f*)(out + b * 8);
    v4f o0 = {O00.re, O00.im, O01.re, O01.im};
    v4f o1 = {O10.re, O10.im, O11.re, O11.im};
    __builtin_nontemporal_store(o0, &ov[0]);
    __builtin_nontemporal_store(o1, &ov[1]);
}

extern "C" void kernel_launch(void* const* d_in, const int* in_sizes, int n_in,
                              void* d_out, int out_size, void* d_ws, size_t ws_size,
                              hipStream_t stream) {
    // setup_inputs order: rho (complex64 [B,2,2]), x (f32 [B,48]),
    //                     w (f32 [48]), theta (f32 [48])
    const float* rho   = (const float*)d_in[0];
    const float* x     = (const float*)d_in[1];
    const float* wv    = (const float*)d_in[2];
    const float* theta = (const float*)d_in[3];
    float* out = (float*)d_out;

    const int B = in_sizes[1] / 48;
    const int block = 256;
    const int grid = (B + block - 1) / block;
    qlayer_su2_kernel<<<grid, block, 0, stream>>>(rho, x, wv, theta, out, B);
}